// RefineSegmentationMultiChannel_52072183496948
// MI455X (gfx1250) — compile-verified
//
#include <hip/hip_runtime.h>
#include <math.h>

// ---------------- problem constants ----------------
#define C_CH   4
#define H_DIM  192
#define W_DIM  192
#define Z_DIM  160
#define HW     (H_DIM * W_DIM)                 // 36864
#define VOL    ((size_t)HW * Z_DIM)            // 5898240 per channel
#define TOTAL  ((size_t)C_CH * VOL)            // 23592960
#define MAXK   33                               // ks = 2*ceil(2*sigma)+1, sigma clamped to 8
#define NB     512                              // partial-reduction blocks per channel
#define PAD    16                               // z halo (>= max half-kernel)
#define PITCH  196                              // LDS row pitch: mult of 4 (b128 align), conflict-free

typedef float v2f __attribute__((ext_vector_type(2)));
typedef float v8f __attribute__((ext_vector_type(8)));

// ---------------- device-side control block in workspace ----------------
struct Ctl {
  float    psum[C_CH][NB];
  float    psq [C_CH][NB];
  unsigned hist[C_CH][4096];
  float    mean[C_CH], var[C_CH], alpha[C_CH], sigma[C_CH];
  int      ks[C_CH], half[C_CH];
  unsigned remain[C_CH];       // remaining rank (1-indexed) within prefix group
  unsigned prefix[C_CH];       // known high bits of the k-th smallest element
  float    thr[C_CH];
  float    tinv[C_CH];         // 1/exp(u[c])
  float    g[C_CH][MAXK];      // normalized 1D gaussian taps
};

// ---------------- pass 1: per-block partial sum / sumsq ----------------
__global__ void reduce_partial(const float* __restrict__ vol, Ctl* __restrict__ ctl) {
  const int ch  = blockIdx.y;
  const int tid = threadIdx.x;
  const float* p = vol + (size_t)ch * VOL;
  float s = 0.f, q = 0.f;
  const size_t stride = (size_t)gridDim.x * blockDim.x;
  for (size_t i = (size_t)blockIdx.x * blockDim.x + tid; i < VOL; i += stride) {
    float v = p[i];
    s += v;
    q += v * v;
  }
  __shared__ float ss[256], qq[256];
  ss[tid] = s; qq[tid] = q;
  __syncthreads();
  for (int o = 128; o > 0; o >>= 1) {
    if (tid < o) { ss[tid] += ss[tid + o]; qq[tid] += qq[tid + o]; }
    __syncthreads();
  }
  if (tid == 0) {
    ctl->psum[ch][blockIdx.x] = ss[0];
    ctl->psq [ch][blockIdx.x] = qq[0];
  }
}

// ---------------- pass 2: final reduce + MLP + gaussian taps (single block) ----------------
__global__ void stats_mlp(Ctl* __restrict__ ctl,
                          const float* __restrict__ w1, const float* __restrict__ b1,
                          const float* __restrict__ w2, const float* __restrict__ b2,
                          const float* __restrict__ u) {
  __shared__ float red[256];
  const int tid = threadIdx.x;
  for (int ch = 0; ch < C_CH; ++ch) {
    float s = 0.f;
    for (int i = tid; i < NB; i += 256) s += ctl->psum[ch][i];
    red[tid] = s; __syncthreads();
    for (int o = 128; o > 0; o >>= 1) { if (tid < o) red[tid] += red[tid + o]; __syncthreads(); }
    if (tid == 0) ctl->mean[ch] = red[0] / (float)VOL;
    __syncthreads();
    float q = 0.f;
    for (int i = tid; i < NB; i += 256) q += ctl->psq[ch][i];
    red[tid] = q; __syncthreads();
    for (int o = 128; o > 0; o >>= 1) { if (tid < o) red[tid] += red[tid + o]; __syncthreads(); }
    if (tid == 0) {
      float m = ctl->mean[ch];
      ctl->var[ch] = (red[0] - (float)VOL * m * m) / (float)(VOL - 1);  // unbiased
    }
    __syncthreads();
  }
  if (tid < C_CH) {
    const int ch = tid;
    const float mean = ctl->mean[ch], var = ctl->var[ch];
    float h[16];
    #pragma unroll
    for (int j = 0; j < 16; ++j) {
      float t = w1[2 * j] * mean + w1[2 * j + 1] * var + b1[j];
      h[j] = t > 0.f ? t : 0.f;
    }
    float o0 = b2[0], o1 = b2[1];
    #pragma unroll
    for (int j = 0; j < 16; ++j) { o0 += w2[j] * h[j]; o1 += w2[16 + j] * h[j]; }
    float alpha = 1.f / (1.f + expf(-o0));
    float sg    = expf(o1);
    sg = fminf(fmaxf(sg, 1e-3f), 8.0f);                 // keeps ks <= MAXK
    int ksv = 2 * (int)ceilf(2.f * sg) + 1;
    if (ksv > MAXK) ksv = MAXK;
    if (ksv < 1)    ksv = 1;
    const int hf = ksv >> 1;
    ctl->alpha[ch] = alpha; ctl->sigma[ch] = sg;
    ctl->ks[ch] = ksv;      ctl->half[ch]  = hf;
    long long k = (long long)((double)alpha * (double)VOL);
    if (k < 1) k = 1;
    if (k > (long long)VOL - 1) k = (long long)VOL - 1;
    ctl->remain[ch] = (unsigned)k;                      // 1-indexed rank
    ctl->prefix[ch] = 0u;
    ctl->tinv[ch]   = expf(-u[ch]);
    // normalized 1D taps (3D normalized kernel == outer product of normalized 1D)
    float sum = 0.f;
    for (int i = 0; i < MAXK; ++i) {
      float x  = (float)(i - hf);
      float gv = (i < ksv) ? expf(-(x * x) / (2.f * sg * sg)) : 0.f;
      ctl->g[ch][i] = gv; sum += gv;
    }
    for (int i = 0; i < MAXK; ++i) ctl->g[ch][i] /= sum;
  }
  __syncthreads();
  // zero histograms for level-0 pass
  unsigned* hflat = &ctl->hist[0][0];
  for (int i = tid; i < C_CH * 4096; i += 256) hflat[i] = 0u;
}

// ---------------- radix-select histogram pass ----------------
__global__ void hist_pass(const float* __restrict__ vol, Ctl* __restrict__ ctl,
                          int condShift, int binShift, int nbins) {
  __shared__ unsigned h[4096];
  const int ch = blockIdx.y;
  for (int i = threadIdx.x; i < nbins; i += blockDim.x) h[i] = 0u;
  __syncthreads();
  const unsigned pfx = ctl->prefix[ch];
  const float* p = vol + (size_t)ch * VOL;
  const unsigned mask = (unsigned)(nbins - 1);
  const size_t stride = (size_t)gridDim.x * blockDim.x;
  for (size_t i = (size_t)blockIdx.x * blockDim.x + threadIdx.x; i < VOL; i += stride) {
    __builtin_prefetch(p + i + stride, 0, 1);           // global_prefetch_b8
    unsigned b = __float_as_uint(p[i]);                 // vals in [0,1): bit order == value order
    bool ok = (condShift >= 32) || ((b >> condShift) == (pfx >> condShift));
    if (ok) atomicAdd(&h[(b >> binShift) & mask], 1u);
  }
  __syncthreads();
  for (int i = threadIdx.x; i < nbins; i += blockDim.x)
    if (h[i]) atomicAdd(&ctl->hist[ch][i], h[i]);       // integer atomics: deterministic
}

// ---------------- radix-select scan (single block) ----------------
__global__ void select_pass(Ctl* __restrict__ ctl, int binShift, int nbins, int last) {
  if (threadIdx.x < C_CH) {
    const int ch = threadIdx.x;
    unsigned rem = ctl->remain[ch];
    unsigned cum = 0;
    int bsel = nbins - 1;
    for (int b = 0; b < nbins; ++b) {
      unsigned c2 = cum + ctl->hist[ch][b];
      if (rem <= c2) { bsel = b; break; }
      cum = c2;
    }
    ctl->remain[ch] = rem - cum;
    unsigned pfx = ctl->prefix[ch] | ((unsigned)bsel << binShift);
    ctl->prefix[ch] = pfx;
    if (last) ctl->thr[ch] = __uint_as_float(pfx);      // exact k-th smallest element
  }
  __syncthreads();
  unsigned* hflat = &ctl->hist[0][0];
  for (int i = threadIdx.x; i < C_CH * 4096; i += blockDim.x) hflat[i] = 0u;
}

// ---------------- separable gaussian: strided passes (H, W) ----------------
__global__ void conv_line(const float* __restrict__ in, float* __restrict__ out,
                          const Ctl* __restrict__ ctl, int SD, int LD) {
  __shared__ float gs[C_CH][MAXK];
  __shared__ int   kss[C_CH], hfs[C_CH];
  if (threadIdx.x < C_CH * MAXK) {
    int ch = threadIdx.x / MAXK, i = threadIdx.x % MAXK;
    gs[ch][i] = ctl->g[ch][i];
  }
  if (threadIdx.x < C_CH) { kss[threadIdx.x] = ctl->ks[threadIdx.x]; hfs[threadIdx.x] = ctl->half[threadIdx.x]; }
  __syncthreads();
  const size_t stride = (size_t)gridDim.x * blockDim.x;
  for (size_t e = (size_t)blockIdx.x * blockDim.x + threadIdx.x; e < TOTAL; e += stride) {
    __builtin_prefetch(in + e + stride, 0, 1);
    const int ch  = (int)(e / VOL);
    const size_t rem = e % VOL;
    const int pos = (int)((rem / (unsigned)SD) % (unsigned)LD);
    const int ksv = kss[ch], hf = hfs[ch];
    const float* base = in + (e - (size_t)pos * SD);    // line start, other coords fixed
    float acc = 0.f;
    for (int d = 0; d < ksv; ++d) {
      int sp = pos + d - hf;                            // zero padding outside [0, LD)
      if (sp >= 0 && sp < LD) acc += gs[ch][d] * base[(size_t)sp * SD];
    }
    out[e] = acc;
  }
}

// ---------------- separable gaussian Z pass: async-LDS staging + V_WMMA_F32_16X16X4_F32 ----------------
// out(16 rows x 16 z) = sum_k A(16 rows x 4 in-z) * B(4 x 16), B[k][n] = g[k-n] (banded).
// Each wave owns a 16-row tile; its 16x160 block is contiguous in global memory and is
// staged into LDS (pitch 196, +-16 zero halo) via global_load_async_to_lds_b128 (ASYNCcnt).
__global__ void conv_z_wmma(const float* __restrict__ in, float* __restrict__ out,
                            const Ctl* __restrict__ ctl) {
  __shared__ float gs[MAXK];
  __shared__ float tile[4][16 * PITCH];                 // 50176 B
  const int ch = (int)(((size_t)blockIdx.x * 64) / HW); // 4 waves x 16 rows; HW % 64 == 0
  if (threadIdx.x < MAXK) gs[threadIdx.x] = ctl->g[ch][threadIdx.x];
  const int ksv = ctl->ks[ch];
  const int hf  = ksv >> 1;
  const int wave = threadIdx.x >> 5, lane = threadIdx.x & 31;
  const size_t row0 = ((size_t)blockIdx.x * 4 + wave) * 16;   // global flattened row
  const float* gbase = in  + row0 * Z_DIM;
  float*       obase = out + row0 * Z_DIM;
  float* lw = &tile[wave][0];

  // zero z-halos: per row dwords [0,PAD) and [PAD+Z, PITCH)  (disjoint from async interior)
  for (int i = lane; i < 16 * (PAD + (PITCH - PAD - Z_DIM)); i += 32) {
    int r = i / 36, c = i % 36;
    int col = (c < PAD) ? c : (c + Z_DIM);
    lw[r * PITCH + col] = 0.f;
  }

  // async-stage interior: 16 rows x 160 dwords; rows are multiples of 4 dwords so each
  // b128 chunk stays within one row; LDS dest (r*196+16+zo)*4 is 16B aligned.
  const unsigned ldsb = (unsigned)(uintptr_t)lw;        // low 32 bits of flat ptr == LDS offset
  #pragma unroll 4
  for (int it = 0; it < 20; ++it) {
    int q  = (it * 32 + lane) * 4;                      // dword index in contiguous 2560-dword block
    int r  = q / Z_DIM;
    int zo = q - r * Z_DIM;
    unsigned laddr = ldsb + (unsigned)(r * PITCH + PAD + zo) * 4u;
    const float* src = gbase + q;
    asm volatile("global_load_async_to_lds_b128 %0, %1, off"
                 :: "v"(laddr), "v"(src) : "memory");
  }
  __syncthreads();                                      // gs (written by threads 0..32) visible to all

  // hoist banded B chunks into registers: invariant across all z tiles
  const int m    = lane & 15;                           // A: M row / B: N col / D: z-in-tile
  const int kofs = (lane >> 4) * 2;                     // K pair selected by half-wave
  const int Kw   = 15 + ksv;                            // input window = 16 + ks - 1
  v2f B[12];
  #pragma unroll
  for (int t = 0; t < 12; ++t) {
    int d = t * 4 + kofs - m;
    B[t].x = (d     >= 0 && d     < ksv) ? gs[d]     : 0.f;
    B[t].y = (d + 1 >= 0 && d + 1 < ksv) ? gs[d + 1] : 0.f;
  }

  asm volatile("s_wait_asynccnt 0x0" ::: "memory");     // staged tile resident in LDS

  const float* arow = lw + m * PITCH;                   // bank-conflict-free: (4m+zi)%64 distinct
  for (int zt = 0; zt < Z_DIM; zt += 16) {
    const int zs = zt - hf + PAD;                       // >= 0 by construction
    // batch all A-chunk LDS loads first so DScnt latency overlaps across chunks,
    // then run the WMMA chain back-to-back (single wait instead of 12 serialized ones)
    v2f a[12];
    #pragma unroll
    for (int t = 0; t < 12; ++t) {
      if (t * 4 < Kw) {                                 // wave-uniform: EXEC stays all-ones
        a[t].x = arow[zs + t * 4 + kofs];
        a[t].y = arow[zs + t * 4 + kofs + 1];
      }
    }
    v8f acc = {};
    #pragma unroll
    for (int t = 0; t < 12; ++t) {
      if (t * 4 < Kw) {                                 // wave-uniform: EXEC stays all-ones
        acc = __builtin_amdgcn_wmma_f32_16x16x4_f32(false, a[t], false, B[t], (short)0, acc, false, false);
      }
    }
    // C/D 16x16 f32: lanes 0-15 hold M=0..7 in v[0..7], lanes 16-31 hold M=8..15; N = lane&15
    const int rb = (lane < 16) ? 0 : 8;
    #pragma unroll
    for (int r = 0; r < 8; ++r)
      obase[(size_t)(rb + r) * Z_DIM + zt + m] = acc[r];
  }
}

// ---------------- fused threshold blend + temperature scaling ----------------
__global__ void finalize_out(const float* __restrict__ vol, float* __restrict__ io,
                             const Ctl* __restrict__ ctl) {
  __shared__ float thr[C_CH], ti[C_CH];
  if (threadIdx.x < C_CH) { thr[threadIdx.x] = ctl->thr[threadIdx.x]; ti[threadIdx.x] = ctl->tinv[threadIdx.x]; }
  __syncthreads();
  const float eps = 1e-6f;
  const size_t stride = (size_t)gridDim.x * blockDim.x;
  for (size_t e = (size_t)blockIdx.x * blockDim.x + threadIdx.x; e < TOTAL; e += stride) {
    const int ch = (int)(e / VOL);
    const float v  = vol[e];
    const float sm = io[e];
    const float r  = (v >= thr[ch]) ? sm : v;           // mask*sm + (1-mask)*v
    const float p  = fminf(fmaxf(r, eps), 1.f - eps);
    const float lg = logf(p / (1.f - p));
    io[e] = 1.f / (1.f + expf(-lg * ti[ch]));
  }
}

// ---------------- launch ----------------
extern "C" void kernel_launch(void* const* d_in, const int* in_sizes, int n_in,
                              void* d_out, int out_size, void* d_ws, size_t ws_size,
                              hipStream_t stream) {
  (void)in_sizes; (void)n_in; (void)out_size; (void)ws_size;
  const float* prob = (const float*)d_in[0];
  const float* w1   = (const float*)d_in[1];
  const float* b1   = (const float*)d_in[2];
  const float* w2   = (const float*)d_in[3];
  const float* b2   = (const float*)d_in[4];
  const float* u    = (const float*)d_in[5];
  float* out = (float*)d_out;

  Ctl* ctl = (Ctl*)d_ws;
  const size_t ctlBytes = (sizeof(Ctl) + 255) & ~(size_t)255;
  float* tmpA = (float*)((char*)d_ws + ctlBytes);       // one ping volume (94.4 MB)

  // stats -> device-resident alpha/sigma/ks/taps/rank
  reduce_partial<<<dim3(NB, C_CH), 256, 0, stream>>>(prob, ctl);
  stats_mlp<<<1, 256, 0, stream>>>(ctl, w1, b1, w2, b2, u);

  // exact k-th smallest via 3-level radix select (12 + 12 + 8 bits)
  hist_pass<<<dim3(240, C_CH), 256, 0, stream>>>(prob, ctl, 32, 20, 4096);
  select_pass<<<1, 256, 0, stream>>>(ctl, 20, 4096, 0);
  hist_pass<<<dim3(240, C_CH), 256, 0, stream>>>(prob, ctl, 20, 8, 4096);
  select_pass<<<1, 256, 0, stream>>>(ctl, 8, 4096, 0);
  hist_pass<<<dim3(240, C_CH), 256, 0, stream>>>(prob, ctl, 8, 0, 256);
  select_pass<<<1, 256, 0, stream>>>(ctl, 0, 256, 1);

  // separable gaussian: H pass -> d_out, W pass -> tmpA, Z pass (WMMA) -> d_out
  conv_line<<<2048, 256, 0, stream>>>(prob, out, ctl, W_DIM * Z_DIM, H_DIM);
  conv_line<<<2048, 256, 0, stream>>>(out, tmpA, ctl, Z_DIM, W_DIM);
  conv_z_wmma<<<(C_CH * HW / 16) / 4, 128, 0, stream>>>(tmpA, out, ctl);

  // threshold blend + temperature scaling (elementwise in-place on d_out)
  finalize_out<<<2048, 256, 0, stream>>>(prob, out, ctl);
}